// GraphSAGE_10694468567472
// MI455X (gfx1250) — compile-verified
//
#include <hip/hip_runtime.h>
#include <hip/hip_bf16.h>

#define D_FEAT 128

typedef float v2f __attribute__((ext_vector_type(2)));
typedef float v8f __attribute__((ext_vector_type(8)));

// ---------------------------------------------------------------------------
// Zero a float buffer (grid-stride).
// ---------------------------------------------------------------------------
__global__ void __launch_bounds__(256) zero_f32(float* __restrict__ p, size_t n) {
    size_t i = (size_t)blockIdx.x * blockDim.x + threadIdx.x;
    size_t stride = (size_t)gridDim.x * blockDim.x;
    for (; i < n; i += stride) p[i] = 0.0f;
}

// ---------------------------------------------------------------------------
// deg[dst[e]] += 1.0f  (one thread per edge)
// ---------------------------------------------------------------------------
__global__ void __launch_bounds__(256) degree_kernel(const int* __restrict__ dst,
                                                     float* __restrict__ deg, int E) {
    int e = blockIdx.x * blockDim.x + threadIdx.x;
    if (e < E) atomicAdd(&deg[dst[e]], 1.0f);
}

// ---------------------------------------------------------------------------
// agg[dst[e], :] += feat[src[e], :]
// One wave32 per edge; each lane moves float4 (32 * 16B = 512B = one row).
// Edge id is wave-uniform -> coherent EXEC, coalesced 512B gather.
// Atomics land in L2 (agg is 51MB, L2 is 192MB).
// ---------------------------------------------------------------------------
__global__ void __launch_bounds__(256) scatter_kernel(const float* __restrict__ feat,
                                                      const int* __restrict__ src,
                                                      const int* __restrict__ dst,
                                                      float* __restrict__ agg, int E) {
    unsigned gid = blockIdx.x * blockDim.x + threadIdx.x;
    unsigned e = gid >> 5;           // wave-uniform
    if (e >= (unsigned)E) return;
    unsigned lane = gid & 31u;
    int s = src[e];
    int d = dst[e];
    const float4 v = *(const float4*)(feat + (size_t)s * D_FEAT + lane * 4);
    float* o = agg + (size_t)d * D_FEAT + lane * 4;
    atomicAdd(o + 0, v.x);
    atomicAdd(o + 1, v.y);
    atomicAdd(o + 2, v.z);
    atomicAdd(o + 3, v.w);
}

// ---------------------------------------------------------------------------
// out[m,n] = act( sum_k (agg[m,k]/max(deg[m],1)) * Wl[n,k]
//               + sum_k  xin[m,k]              * Wr[n,k]  + bl[n] )
//
// One wave per 16x16 output tile, K=128 via 32x V_WMMA_F32_16X16X4_F32 per
// GEMM. Two independent accumulators so the two WMMA chains interleave.
//
// 16x16x4 f32 fragment layout (ISA 7.12.2):
//   A (16x4):  lane l<16 row m=l holds K={k,k+1}; lanes 16..31 hold K={k+2,k+3}
//   B (4x16):  lane l<16 col n=l holds K={k,k+1}; lanes 16..31 hold K={k+2,k+3}
//   C/D:       VGPR r, lane<16 -> (M=r,   N=lane); lane>=16 -> (M=r+8, N=lane-16)
// Both A and B fragments are contiguous float2 loads (B is W^T: W[n][k..k+1]).
// ---------------------------------------------------------------------------
template <int N_DIM, bool RELU>
__global__ void __launch_bounds__(256) sage_gemm(
    const float* __restrict__ agg, const float* __restrict__ deg,
    const float* __restrict__ xin, const float* __restrict__ Wl,
    const float* __restrict__ bl,  const float* __restrict__ Wr,
    float* __restrict__ out, int M) {
    constexpr int K = D_FEAT;
    constexpr int tilesN = N_DIM / 16;

    const int lane   = threadIdx.x & 31;
    const int wave   = threadIdx.x >> 5;
    const int tileId = blockIdx.x * (blockDim.x >> 5) + wave;  // wave-uniform
    const int totalTiles = (M / 16) * tilesN;
    if (tileId >= totalTiles) return;  // whole wave exits -> EXEC all-1s for WMMA

    const int tm = (tileId / tilesN) * 16;
    const int tn = (tileId % tilesN) * 16;

    const int half = lane >> 4;      // 0: lanes 0-15, 1: lanes 16-31
    const int l15  = lane & 15;
    const int kOff = half * 2;       // upper half holds K+2,K+3

    const int mA = tm + l15;         // A-fragment row for this lane
    const int nB = tn + l15;         // B-fragment column for this lane

    const float invdeg = 1.0f / fmaxf(deg[mA], 1.0f);

    const float* aRow  = agg + (size_t)mA * K + kOff;
    const float* xRow  = xin + (size_t)mA * K + kOff;
    const float* wlRow = Wl  + (size_t)nB * K + kOff;
    const float* wrRow = Wr  + (size_t)nB * K + kOff;

    v8f acc0 = {};
    v8f acc1 = {};
#pragma unroll
    for (int k = 0; k < K; k += 4) {
        v2f a0 = *(const v2f*)(aRow + k);
        a0 *= invdeg;                               // fold mean-normalization into A
        v2f b0 = *(const v2f*)(wlRow + k);
        acc0 = __builtin_amdgcn_wmma_f32_16x16x4_f32(
            false, a0, false, b0, (short)0, acc0, false, false);

        v2f a1 = *(const v2f*)(xRow + k);
        v2f b1 = *(const v2f*)(wrRow + k);
        acc1 = __builtin_amdgcn_wmma_f32_16x16x4_f32(
            false, a1, false, b1, (short)0, acc1, false, false);
    }

    const float bias  = bl[tn + l15];
    const int   mBase = tm + half * 8;
#pragma unroll
    for (int r = 0; r < 8; ++r) {
        float v = acc0[r] + acc1[r] + bias;
        if (RELU) v = fmaxf(v, 0.0f);
        out[(size_t)(mBase + r) * N_DIM + tn + l15] = v;
    }
}

// ---------------------------------------------------------------------------
// Orchestration. Inputs (setup_inputs order):
//   0: x [N,128] f32     1: edge_index [2,E] int     2: Wl1 [128,128]
//   3: bl1 [128]         4: Wr1 [128,128]            5: Wl2 [64,128]
//   6: bl2 [64]          7: Wr2 [64,128]
// Output: [N,64] f32.
// Workspace: agg [N*128] | deg [N] | h [N*128]  (~103 MB)
// ---------------------------------------------------------------------------
extern "C" void kernel_launch(void* const* d_in, const int* in_sizes, int n_in,
                              void* d_out, int out_size, void* d_ws, size_t ws_size,
                              hipStream_t stream) {
    const float* x   = (const float*)d_in[0];
    const int*   ei  = (const int*)d_in[1];
    const float* Wl1 = (const float*)d_in[2];
    const float* bl1 = (const float*)d_in[3];
    const float* Wr1 = (const float*)d_in[4];
    const float* Wl2 = (const float*)d_in[5];
    const float* bl2 = (const float*)d_in[6];
    const float* Wr2 = (const float*)d_in[7];

    const int N = in_sizes[0] / D_FEAT;  // 100000
    const int E = in_sizes[1] / 2;       // 1600000

    const int* src = ei;
    const int* dst = ei + E;

    float* agg = (float*)d_ws;                    // N*128
    float* deg = agg + (size_t)N * D_FEAT;        // N
    float* h   = deg + N;                         // N*128

    const int scatterBlocks = (int)(((long long)E * 32 + 255) / 256);

    // ---- Layer 1 ----
    zero_f32<<<2048, 256, 0, stream>>>(agg, (size_t)N * D_FEAT + N);  // agg + deg
    degree_kernel<<<(E + 255) / 256, 256, 0, stream>>>(dst, deg, E);
    scatter_kernel<<<scatterBlocks, 256, 0, stream>>>(x, src, dst, agg, E);
    {
        int tiles = (N / 16) * (D_FEAT / 16);
        sage_gemm<D_FEAT, true><<<(tiles + 7) / 8, 256, 0, stream>>>(
            agg, deg, x, Wl1, bl1, Wr1, h, N);
    }

    // ---- Layer 2 (same edges -> reuse deg) ----
    zero_f32<<<2048, 256, 0, stream>>>(agg, (size_t)N * D_FEAT);
    scatter_kernel<<<scatterBlocks, 256, 0, stream>>>(h, src, dst, agg, E);
    {
        int tiles = (N / 16) * (64 / 16);
        sage_gemm<64, false><<<(tiles + 7) / 8, 256, 0, stream>>>(
            agg, deg, h, Wl2, bl2, Wr2, (float*)d_out, N);
    }
}